// RecurrentLinearAttention_2688649527589
// MI455X (gfx1250) — compile-verified
//
#include <hip/hip_runtime.h>

// CDNA5 / gfx1250: wave32, WMMA f32 16x16x4
typedef float v2f __attribute__((ext_vector_type(2)));
typedef float v8f __attribute__((ext_vector_type(8)));

#define RLA_EPS 1e-6f

__device__ __forceinline__ float elu_p1(float x) {
    // elu(x)+1 = x+1 (x>0) else exp(x)
    return x > 0.0f ? x + 1.0f : __expf(x);
}

// One wave (32 lanes) per (n,h) pair. D = M = 64.
// Layout in/out is fully flat: pair p owns q/k/v/Zi[p*64 .. +63], Si[p*4096 .. +4095].
__global__ __launch_bounds__(256)
void rla_step_kernel(const float* __restrict__ q,
                     const float* __restrict__ k,
                     const float* __restrict__ v,
                     const float* __restrict__ Si,
                     const float* __restrict__ Zi,
                     float* __restrict__ outV,    // (NH, 64)
                     float* __restrict__ outSi,   // (NH, 64, 64)
                     float* __restrict__ outZi)   // (NH, 64)
{
    const int lane = threadIdx.x & 31;
    const int wave = threadIdx.x >> 5;
    const int p    = blockIdx.x * 8 + wave;      // pair index n*H + h

    __shared__ float lds_qf[8][64];
    __shared__ float lds_kf[8][64];
    __shared__ float lds_v [8][64];

    const size_t vec_base = (size_t)p * 64;
    const size_t si_base  = (size_t)p * 4096;

    // ---- Phase 1: feature maps, Zi_new, normalizer Z -------------------
    float Zscale;
    {
        const int e = 2 * lane;                  // lane owns elements e, e+1
        v2f q2 = *(const v2f*)(q  + vec_base + e);
        v2f k2 = *(const v2f*)(k  + vec_base + e);
        v2f w2 = *(const v2f*)(v  + vec_base + e);
        v2f z2 = *(const v2f*)(Zi + vec_base + e);

        v2f qf, kf, zn;
        qf.x = elu_p1(q2.x); qf.y = elu_p1(q2.y);
        kf.x = elu_p1(k2.x); kf.y = elu_p1(k2.y);
        zn.x = z2.x + kf.x;  zn.y = z2.y + kf.y;

        *(v2f*)(outZi + vec_base + e) = zn;      // Zi_new out
        *(v2f*)&lds_qf[wave][e] = qf;
        *(v2f*)&lds_kf[wave][e] = kf;
        *(v2f*)&lds_v [wave][e] = w2;

        // s = dot(Qf, Zi_new) over all 64 elements: wave32 butterfly
        float s = qf.x * zn.x + qf.y * zn.y;
        s += __shfl_xor(s, 1);
        s += __shfl_xor(s, 2);
        s += __shfl_xor(s, 4);
        s += __shfl_xor(s, 8);
        s += __shfl_xor(s, 16);
        Zscale = 1.0f / (s + RLA_EPS);
    }
    __syncthreads();   // make LDS staging visible (uniform, all lanes reach it)

    // ---- Phase 2: stream Si: rank-1 update + WMMA contraction ----------
    // WMMA 16x16x4 f32 operand layout (wave32):
    //   A (16x4): lane 0-15 rows, half selects K{0,1}/K{2,3}; all rows = Qf chunk
    //   B (4x16): lane&15 = column m, half selects K{0,1}/K{2,3}
    const int half = lane >> 4;
    const int mcol = lane & 15;

    float vm[4];
    #pragma unroll
    for (int mt = 0; mt < 4; ++mt)
        vm[mt] = lds_v[wave][mcol + 16 * mt];

    v8f acc[4] = {};   // four 16x16 C tiles covering m = 0..63

    for (int c = 0; c < 16; ++c) {
        const int d0 = 4 * c;
        const int r0 = d0 + 2 * half;            // this lane's first Si row
        v2f a  = *(const v2f*)&lds_qf[wave][r0]; // A operand (rows broadcast)
        v2f kf = *(const v2f*)&lds_kf[wave][r0];

        const size_t rowidx = si_base + (size_t)r0 * 64 + mcol;
        #pragma unroll
        for (int mt = 0; mt < 4; ++mt) {
            const size_t idx = rowidx + 16 * mt;
            float s0 = Si[idx];                  // row r0,   col m
            float s1 = Si[idx + 64];             // row r0+1, col m
            v2f b;
            b.x = fmaf(kf.x, vm[mt], s0);        // Si_new = Si + Kf[d]*v[m]
            b.y = fmaf(kf.y, vm[mt], s1);
            outSi[idx]      = b.x;               // Si_new out
            outSi[idx + 64] = b.y;
            // acc += A(Qf bcast) x B(Si_new chunk): every C row = partial V
            acc[mt] = __builtin_amdgcn_wmma_f32_16x16x4_f32(
                false, a, false, b, (short)0, acc[mt], false, false);
        }
    }

    // C row 0 lives in element 0 of the v8f on lanes 0..15 (N = lane)
    if (lane < 16) {
        #pragma unroll
        for (int mt = 0; mt < 4; ++mt)
            outV[vec_base + 16 * mt + lane] = Zscale * acc[mt][0];
    }
}

extern "C" void kernel_launch(void* const* d_in, const int* in_sizes, int n_in,
                              void* d_out, int out_size, void* d_ws, size_t ws_size,
                              hipStream_t stream) {
    (void)n_in; (void)out_size; (void)d_ws; (void)ws_size;
    const float* q  = (const float*)d_in[0];
    const float* k  = (const float*)d_in[1];
    const float* v  = (const float*)d_in[2];
    const float* Si = (const float*)d_in[3];
    const float* Zi = (const float*)d_in[4];

    const size_t NH = (size_t)in_sizes[0] / 64;  // N*H (D=64)

    float* out   = (float*)d_out;
    float* outV  = out;                           // (NH,64)
    float* outSi = out + NH * 64;                 // (NH,64,64)
    float* outZi = out + NH * 64 + NH * 4096;     // (NH,64)

    dim3 grid((unsigned)(NH / 8)), block(256);
    hipLaunchKernelGGL(rla_step_kernel, grid, block, 0, stream,
                       q, k, v, Si, Zi, outV, outSi, outZi);
}